// HistogramObserver_79362405695707
// MI455X (gfx1250) — compile-verified
//
#include <hip/hip_runtime.h>
#include <cstdint>

// ---------------- workspace layout (uint32 units) ----------------
#define NB1 2048   // digit 1: abs_bits[30:20]  (11 bits)
#define NB2 2048   // digit 2: abs_bits[19:9]   (11 bits)
#define NB3 512    // digit 3: abs_bits[8:0]    (9 bits)
#define WS_H1 0
#define WS_H2 (WS_H1 + NB1)
#define WS_H3 (WS_H2 + NB2)
#define WS_PREFIX (WS_H3 + NB3)
#define WS_K (WS_PREFIX + 1)
#define WS_MINMONO (WS_K + 1)
#define WS_MINARR (WS_MINMONO + 1)

#define P1_WAVES 8          // 256-thread blocks = 8 wave32s
#define H1_REP 4            // bank-interleaved histogram replicas in pass 1

__device__ __forceinline__ unsigned mono_map(unsigned u) {
  // order-preserving map for signed float bit patterns -> unsigned
  return (u & 0x80000000u) ? ~u : (u | 0x80000000u);
}

// ---------------- init ----------------
__global__ __launch_bounds__(1024) void hobs_init_kernel(unsigned* ws, unsigned k) {
  int i = blockIdx.x * blockDim.x + threadIdx.x;
  if (i < WS_PREFIX) ws[i] = 0u;          // zero all three histograms
  if (i == 0) {
    ws[WS_PREFIX]  = 0u;
    ws[WS_K]       = k;
    ws[WS_MINMONO] = 0xFFFFFFFFu;
  }
}

// ---------------- pass 1: top-digit histogram + min, async-LDS pipelined ----------------
__global__ __launch_bounds__(256) void hobs_pass1_kernel(const float* __restrict__ in,
                                                         long long n,
                                                         unsigned* __restrict__ ghist,
                                                         unsigned* __restrict__ minarr) {
  // replica index interleaved innermost: addr = bin*H1_REP + rep
  // -> same-bin updates from one wave spread over H1_REP consecutive banks,
  //    and a contiguous hot-bin region covers all 64 LDS banks.
  __shared__ unsigned hist[NB1 * H1_REP];
  __shared__ float4   stage[P1_WAVES][2][32];
  __shared__ unsigned bmin;

  const int tid = threadIdx.x;
  for (int b = tid; b < NB1 * H1_REP; b += blockDim.x) hist[b] = 0u;
  if (tid == 0) bmin = 0xFFFFFFFFu;
  __syncthreads();

  const int wave = tid >> 5;
  const int lane = tid & 31;
  const int rep  = lane & (H1_REP - 1);
  const long long nchunks = n >> 7;                 // 128 floats per wave-chunk
  const long long gwave   = (long long)blockIdx.x * P1_WAVES + wave;
  const long long wstride = (long long)gridDim.x * P1_WAVES;

  unsigned mn = 0xFFFFFFFFu;

  // per-lane LDS byte addresses of the two 16B staging slots
  unsigned lds0 = (unsigned)(uintptr_t)&stage[wave][0][lane];
  unsigned lds1 = (unsigned)(uintptr_t)&stage[wave][1][lane];

  long long c = gwave;
  if (c < nchunks) {                                // wave-uniform condition: EXEC all-1s
    unsigned goff = (unsigned)((c << 7) + (long long)(lane << 2)) << 2; // byte offset
    asm volatile("global_load_async_to_lds_b128 %0, %1, %2"
                 :: "v"(lds0), "v"(goff), "s"(in) : "memory");
    int buf = 0;
    for (;;) {
      long long cn = c + wstride;
      if (cn < nchunks) {
        unsigned goff2 = (unsigned)((cn << 7) + (long long)(lane << 2)) << 2;
        asm volatile("global_load_async_to_lds_b128 %0, %1, %2"
                     :: "v"(buf ? lds0 : lds1), "v"(goff2), "s"(in) : "memory");
        // async loads complete in order: <=1 outstanding => older buffer landed
        asm volatile("s_wait_asynccnt 0x1" ::: "memory");
      } else {
        asm volatile("s_wait_asynccnt 0x0" ::: "memory");
      }
      float4 v = stage[wave][buf][lane];
      {
        unsigned u, a;
        u = __float_as_uint(v.x); a = u & 0x7FFFFFFFu;
        atomicAdd(&hist[((a >> 20) << 2) | rep], 1u); mn = min(mn, mono_map(u));
        u = __float_as_uint(v.y); a = u & 0x7FFFFFFFu;
        atomicAdd(&hist[((a >> 20) << 2) | rep], 1u); mn = min(mn, mono_map(u));
        u = __float_as_uint(v.z); a = u & 0x7FFFFFFFu;
        atomicAdd(&hist[((a >> 20) << 2) | rep], 1u); mn = min(mn, mono_map(u));
        u = __float_as_uint(v.w); a = u & 0x7FFFFFFFu;
        atomicAdd(&hist[((a >> 20) << 2) | rep], 1u); mn = min(mn, mono_map(u));
      }
      if (cn >= nchunks) break;
      c = cn; buf ^= 1;
    }
  }

  // tail (n not a multiple of 128)
  const long long tail = nchunks << 7;
  const long long tthreads = (long long)gridDim.x * blockDim.x;
  for (long long i = tail + (long long)blockIdx.x * blockDim.x + tid; i < n; i += tthreads) {
    unsigned u = __float_as_uint(in[i]);
    unsigned a = u & 0x7FFFFFFFu;
    atomicAdd(&hist[((a >> 20) << 2) | rep], 1u);
    mn = min(mn, mono_map(u));
  }

  atomicMin(&bmin, mn);
  __syncthreads();
  for (int b = tid; b < NB1; b += blockDim.x) {
    unsigned cu = hist[b << 2] + hist[(b << 2) | 1] + hist[(b << 2) | 2] + hist[(b << 2) | 3];
    if (cu) atomicAdd(&ghist[b], cu);
  }
  if (tid == 0) minarr[blockIdx.x] = bmin;
}

// ---------------- refinement passes (L2-resident re-reads) ----------------
__global__ __launch_bounds__(256) void hobs_refine_kernel(const float* __restrict__ in,
                                                          long long n,
                                                          const unsigned* __restrict__ state_prefix,
                                                          unsigned* __restrict__ ghist,
                                                          int cmp_shift, int dig_shift,
                                                          unsigned dig_mask, int nbins) {
  __shared__ unsigned hist[NB1];  // max bins = 2048
  const int tid = threadIdx.x;
  for (int b = tid; b < nbins; b += blockDim.x) hist[b] = 0u;
  __syncthreads();

  const unsigned prefix = *state_prefix;   // uniform scalar load
  const long long nv = n >> 2;
  const long long gstride = (long long)gridDim.x * blockDim.x;
  const float4* in4 = (const float4*)in;

  for (long long i = (long long)blockIdx.x * blockDim.x + tid; i < nv; i += gstride) {
    float4 v = in4[i];
    unsigned a;
    a = __float_as_uint(v.x) & 0x7FFFFFFFu; if ((a >> cmp_shift) == prefix) atomicAdd(&hist[(a >> dig_shift) & dig_mask], 1u);
    a = __float_as_uint(v.y) & 0x7FFFFFFFu; if ((a >> cmp_shift) == prefix) atomicAdd(&hist[(a >> dig_shift) & dig_mask], 1u);
    a = __float_as_uint(v.z) & 0x7FFFFFFFu; if ((a >> cmp_shift) == prefix) atomicAdd(&hist[(a >> dig_shift) & dig_mask], 1u);
    a = __float_as_uint(v.w) & 0x7FFFFFFFu; if ((a >> cmp_shift) == prefix) atomicAdd(&hist[(a >> dig_shift) & dig_mask], 1u);
  }
  for (long long i = (nv << 2) + (long long)blockIdx.x * blockDim.x + tid; i < n; i += gstride) {
    unsigned a = __float_as_uint(in[i]) & 0x7FFFFFFFu;
    if ((a >> cmp_shift) == prefix) atomicAdd(&hist[(a >> dig_shift) & dig_mask], 1u);
  }

  __syncthreads();
  for (int b = tid; b < nbins; b += blockDim.x) {
    unsigned cu = hist[b];
    if (cu) atomicAdd(&ghist[b], cu);
  }
}

// ---------------- digit select (single block) + optional min-reduce / finalize ----------------
__global__ __launch_bounds__(1024) void hobs_scan_kernel(const unsigned* __restrict__ hist, int nbins,
                                                         int dig_bits,
                                                         unsigned* state_prefix, unsigned* state_k,
                                                         int first_scan, const unsigned* __restrict__ minarr,
                                                         int nmin, unsigned* minmono,
                                                         int final_scan,
                                                         const float* __restrict__ minv,
                                                         const float* __restrict__ maxv,
                                                         const int* __restrict__ numflag,
                                                         float* __restrict__ out) {
  __shared__ unsigned sh[1024];
  const int t = threadIdx.x;
  const int bpt = (nbins + 1023) >> 10;
  const int start = t * bpt;

  unsigned lsum = 0u;
  for (int j = 0; j < bpt; ++j) {
    int b = start + j;
    if (b < nbins) lsum += hist[b];
  }
  sh[t] = lsum;
  __syncthreads();
  for (int off = 1; off < 1024; off <<= 1) {
    unsigned v = (t >= off) ? sh[t - off] : 0u;
    __syncthreads();
    sh[t] += v;
    __syncthreads();
  }
  const unsigned excl = sh[t] - lsum;

  const unsigned k = *state_k;
  unsigned cum = excl;
  for (int j = 0; j < bpt; ++j) {
    int b = start + j;
    if (b >= nbins) break;
    unsigned c = hist[b];
    if (c && cum < k && cum + c >= k) {
      unsigned np = ((*state_prefix) << dig_bits) | (unsigned)b;
      *state_prefix = np;
      *state_k = k - cum;
      if (final_scan) {
        float maxcur = __uint_as_float(np);    // exact k-th smallest of |x|
        unsigned mm = *minmono;
        unsigned mbits = (mm & 0x80000000u) ? (mm ^ 0x80000000u) : ~mm;
        float mincur = __uint_as_float(mbits);
        int flag = *numflag;
        float nmax = (flag == 0) ? maxcur : (0.9f * maxv[0] + 0.1f * maxcur);
        float nmin = (flag == 0) ? mincur : (0.9f * minv[0] + 0.1f * mincur);
        out[0] = nmin;
        out[1] = nmax;
      }
    }
    cum += c;
  }

  if (first_scan) {
    unsigned m = 0xFFFFFFFFu;
    for (int i = t; i < nmin; i += 1024) m = min(m, minarr[i]);
    __syncthreads();
    sh[t] = m;
    __syncthreads();
    for (int off = 512; off > 0; off >>= 1) {
      if (t < off) sh[t] = min(sh[t], sh[t + off]);
      __syncthreads();
    }
    if (t == 0) *minmono = sh[0];
  }
}

extern "C" void kernel_launch(void* const* d_in, const int* in_sizes, int n_in,
                              void* d_out, int out_size, void* d_ws, size_t ws_size,
                              hipStream_t stream) {
  const float* in      = (const float*)d_in[0];
  const float* minv    = (const float*)d_in[1];
  const float* maxv    = (const float*)d_in[2];
  const int*   numflag = (const int*)d_in[3];
  float*       out     = (float*)d_out;
  unsigned*    ws      = (unsigned*)d_ws;

  const long long n = (long long)in_sizes[0];
  long long kll = (long long)(0.9999 * (double)n);   // matches Python int() truncation
  if (kll < 1) kll = 1;
  if (kll > n) kll = n;
  const unsigned k = (unsigned)kll;

  int p1_blocks = 4096;
  {
    size_t need = (size_t)(WS_MINARR + p1_blocks) * sizeof(unsigned);
    if (ws_size < need) {
      long long avail = (long long)(ws_size / sizeof(unsigned)) - WS_MINARR;
      if (avail < 1) avail = 1;
      if (avail > 4096) avail = 4096;
      p1_blocks = (int)avail;
    }
  }

  hobs_init_kernel<<<(WS_PREFIX + 1023) / 1024, 1024, 0, stream>>>(ws, k);

  hobs_pass1_kernel<<<p1_blocks, 256, 0, stream>>>(in, n, ws + WS_H1, ws + WS_MINARR);

  hobs_scan_kernel<<<1, 1024, 0, stream>>>(ws + WS_H1, NB1, 11,
                                           ws + WS_PREFIX, ws + WS_K,
                                           1, ws + WS_MINARR, p1_blocks, ws + WS_MINMONO,
                                           0, minv, maxv, numflag, out);

  hobs_refine_kernel<<<4096, 256, 0, stream>>>(in, n, ws + WS_PREFIX, ws + WS_H2,
                                               20, 9, 0x7FFu, NB2);

  hobs_scan_kernel<<<1, 1024, 0, stream>>>(ws + WS_H2, NB2, 11,
                                           ws + WS_PREFIX, ws + WS_K,
                                           0, nullptr, 0, ws + WS_MINMONO,
                                           0, minv, maxv, numflag, out);

  hobs_refine_kernel<<<4096, 256, 0, stream>>>(in, n, ws + WS_PREFIX, ws + WS_H3,
                                               9, 0, 0x1FFu, NB3);

  hobs_scan_kernel<<<1, 1024, 0, stream>>>(ws + WS_H3, NB3, 9,
                                           ws + WS_PREFIX, ws + WS_K,
                                           0, nullptr, 0, ws + WS_MINMONO,
                                           1, minv, maxv, numflag, out);
}